// ScaledDotProductAttention_84061099917656
// MI455X (gfx1250) — compile-verified
//
#include <hip/hip_runtime.h>
#include <hip/hip_bf16.h>

// Scaled dot product attention, fp32, returns (out[B,N,D], attn[B,N,N]).
// B=16, N=2048, D=64, temperature=8 (scale=1/8).
// One workgroup = one batch x 16 query rows. 256 threads = 8 wave32.
// Matrix math on V_WMMA_F32_16X16X4_F32 (exact fp32, matches reference).
// attn stores + mask loads are non-temporal: the 268MB attn stream must not
// evict the 16MB K/V working set from the 192MB L2 (re-read by every q-tile).

#define ATTN_B 16
#define ATTN_N 2048
#define ATTN_D 64
#define SPITCH 2052   // 2048 + 4 floats pad: kills LDS bank conflicts, stays 16B-aligned

typedef __attribute__((ext_vector_type(2))) float v2f;
typedef __attribute__((ext_vector_type(4))) float v4f;
typedef __attribute__((ext_vector_type(8))) float v8f;
typedef __attribute__((ext_vector_type(4))) unsigned char v4u8;

__launch_bounds__(256, 1)
__global__ void sdpa_kernel(const float* __restrict__ qg,
                            const float* __restrict__ kg,
                            const float* __restrict__ vg,
                            const unsigned char* __restrict__ maskg,
                            float* __restrict__ outg,
                            float* __restrict__ attng)
{
    // 16 rows x 2052 floats of scores/probs; head re-used as O partials later.
    __shared__ float S[16 * SPITCH];            // 131328 B
    __shared__ float pred[16][16];              // partial row reductions
    __shared__ float rowmax[16];
    __shared__ float rowinv[16];

    const int tid  = threadIdx.x;
    const int wave = tid >> 5;          // 0..7
    const int lane = tid & 31;
    const int b    = blockIdx.y;
    const int q0   = blockIdx.x * 16;

    const float* qb = qg + ((size_t)b * ATTN_N + q0) * ATTN_D;
    const float* kb = kg + (size_t)b * ATTN_N * ATTN_D;
    const float* vb = vg + (size_t)b * ATTN_N * ATTN_D;

    // WMMA f32 16x16x4 lane mapping (ISA 7.12.2):
    //  A (16x4):  lane m = lane&15, VGPR0 = K{0 | 2}, VGPR1 = K{1 | 3} (lanes 0-15 | 16-31)
    //  B (4x16):  lane n = lane&15, same K split
    //  C/D (16x16): lane n = lane&15, VGPR r -> row m = r + 8*(lane>>4)
    const int l16 = lane & 15;
    const int ksub = (lane >> 4) * 2;       // 0 for lanes 0-15, 2 for lanes 16-31
    const int mrow_hi = 8 * (lane >> 4);

    // ---------------- Phase 1: S = (Q K^T) / 8 ----------------
    v2f afrag[16];
#pragma unroll
    for (int s = 0; s < 16; ++s) {
        const float* p = qb + (size_t)l16 * ATTN_D + 4 * s + ksub;
        afrag[s] = *(const v2f*)p;          // 8B aligned (even float index)
    }

    for (int t = 0; t < 16; ++t) {
        const int key0 = (wave + t * 8) * 16;   // this wave's key tile
        v8f c = {};
#pragma unroll
        for (int s = 0; s < 16; ++s) {
            const float* p = kb + (size_t)(key0 + l16) * ATTN_D + 4 * s + ksub;
            v2f bfrag = *(const v2f*)p;
            c = __builtin_amdgcn_wmma_f32_16x16x4_f32(
                    false, afrag[s], false, bfrag, (short)0, c, false, false);
        }
#pragma unroll
        for (int r = 0; r < 8; ++r) {
            S[(r + mrow_hi) * SPITCH + key0 + l16] = c[r] * 0.125f;
        }
    }
    __syncthreads();

    // ---------------- Phase 2: mask + row max (float4 / uchar4) ----------------
    const int row = tid >> 4;       // 0..15 (16 threads cooperate per row)
    const int sub = tid & 15;
    const int cbase = sub * 4;      // each thread owns 4 consecutive cols per 64-col step
    const unsigned char* mrow = maskg + ((size_t)b * ATTN_N + q0 + row) * ATTN_N;
    const float NEG_INF = -__builtin_inff();

    float pmax = NEG_INF;
    for (int it = 0; it < 32; ++it) {
        const int c0 = cbase + it * 64;
        v4u8 mm = __builtin_nontemporal_load((const v4u8*)(mrow + c0));
        v4f s = *(v4f*)&S[row * SPITCH + c0];
        if (mm.x) s.x = NEG_INF;
        if (mm.y) s.y = NEG_INF;
        if (mm.z) s.z = NEG_INF;
        if (mm.w) s.w = NEG_INF;
        *(v4f*)&S[row * SPITCH + c0] = s;
        pmax = fmaxf(pmax, fmaxf(fmaxf(s.x, s.y), fmaxf(s.z, s.w)));
    }
    pred[row][sub] = pmax;
    __syncthreads();
    if (tid < 16) {
        float m = NEG_INF;
#pragma unroll
        for (int i = 0; i < 16; ++i) m = fmaxf(m, pred[tid][i]);
        rowmax[tid] = m;
    }
    __syncthreads();

    // ---------------- Phase 3a: exp + row sum ----------------
    const float rmax = rowmax[row];
    const bool dead = (rmax == NEG_INF);    // fully-masked row -> all zeros
    float psum = 0.0f;
    for (int it = 0; it < 32; ++it) {
        const int c0 = cbase + it * 64;
        v4f s = *(v4f*)&S[row * SPITCH + c0];
        v4f e;
        e.x = dead ? 0.0f : __expf(s.x - rmax);   // exp(-inf - m) = 0 handles masked
        e.y = dead ? 0.0f : __expf(s.y - rmax);
        e.z = dead ? 0.0f : __expf(s.z - rmax);
        e.w = dead ? 0.0f : __expf(s.w - rmax);
        *(v4f*)&S[row * SPITCH + c0] = e;
        psum += (e.x + e.y) + (e.z + e.w);
    }
    pred[row][sub] = psum;
    __syncthreads();
    if (tid < 16) {
        float l = 0.0f;
#pragma unroll
        for (int i = 0; i < 16; ++i) l += pred[tid][i];
        rowinv[tid] = (l > 0.0f) ? (1.0f / l) : 0.0f;
    }
    __syncthreads();

    // ---------------- Phase 3b: normalize, write attn (NT b128 stores) ----------------
    const float rinv = rowinv[row];
    float* arow = attng + ((size_t)b * ATTN_N + q0 + row) * ATTN_N;
    for (int it = 0; it < 32; ++it) {
        const int c0 = cbase + it * 64;
        v4f p = *(v4f*)&S[row * SPITCH + c0];
        p.x *= rinv; p.y *= rinv; p.z *= rinv; p.w *= rinv;
        *(v4f*)&S[row * SPITCH + c0] = p;
        __builtin_nontemporal_store(p, (v4f*)(arow + c0));
    }
    __syncthreads();

    // ---------------- Phase 4: O = P V ----------------
    // Each wave accumulates over its private key range [wave*256, wave*256+256).
    v8f o0 = {}, o1 = {}, o2 = {}, o3 = {};
    const int kbase = wave * 256;
    for (int s = 0; s < 64; ++s) {
        const int kk = kbase + 4 * s + ksub;
        v2f a = *(const v2f*)&S[l16 * SPITCH + kk];     // P[m][kk], P[m][kk+1]
        const float* vr0 = vb + (size_t)kk * ATTN_D + l16;
        const float* vr1 = vr0 + ATTN_D;
        {
            v2f bf = { vr0[0],  vr1[0]  };
            o0 = __builtin_amdgcn_wmma_f32_16x16x4_f32(false, a, false, bf, (short)0, o0, false, false);
        }
        {
            v2f bf = { vr0[16], vr1[16] };
            o1 = __builtin_amdgcn_wmma_f32_16x16x4_f32(false, a, false, bf, (short)0, o1, false, false);
        }
        {
            v2f bf = { vr0[32], vr1[32] };
            o2 = __builtin_amdgcn_wmma_f32_16x16x4_f32(false, a, false, bf, (short)0, o2, false, false);
        }
        {
            v2f bf = { vr0[48], vr1[48] };
            o3 = __builtin_amdgcn_wmma_f32_16x16x4_f32(false, a, false, bf, (short)0, o3, false, false);
        }
    }
    __syncthreads();    // everyone done reading S -> safe to alias as Opart

    float* Opart = S;   // 8 waves x (16x64) floats = 32KB, aliases head of S
#pragma unroll
    for (int r = 0; r < 8; ++r) {
        const int m = r + mrow_hi;
        Opart[(wave * 1024) + m * 64 +  0 + l16] = o0[r];
        Opart[(wave * 1024) + m * 64 + 16 + l16] = o1[r];
        Opart[(wave * 1024) + m * 64 + 32 + l16] = o2[r];
        Opart[(wave * 1024) + m * 64 + 48 + l16] = o3[r];
    }
    __syncthreads();

    // Cross-wave reduction + store: 1024 outputs, 4 per thread, coalesced.
    for (int e = tid; e < 1024; e += 256) {
        float acc = 0.0f;
#pragma unroll
        for (int w = 0; w < 8; ++w) acc += Opart[w * 1024 + e];
        outg[((size_t)b * ATTN_N + q0 + (e >> 6)) * ATTN_D + (e & 63)] = acc;
    }
}

extern "C" void kernel_launch(void* const* d_in, const int* in_sizes, int n_in,
                              void* d_out, int out_size, void* d_ws, size_t ws_size,
                              hipStream_t stream) {
    const float* q = (const float*)d_in[0];
    const float* k = (const float*)d_in[1];
    const float* v = (const float*)d_in[2];
    const unsigned char* mask = (const unsigned char*)d_in[3];   // jax bool = 1 byte

    float* out  = (float*)d_out;
    float* attn = out + (size_t)ATTN_B * ATTN_N * ATTN_D;

    dim3 grid(ATTN_N / 16, ATTN_B);   // 128 q-tiles x 16 batches
    sdpa_kernel<<<grid, 256, 0, stream>>>(q, k, v, mask, out, attn);
}